// MarkovModel_62079457296950
// MI455X (gfx1250) — compile-verified
//
#include <hip/hip_runtime.h>
#include <stdint.h>

// Problem constants (fixed by the harness / reference setup_inputs()).
#define K_STATES 4096
#define N_SEQ    8192
#define SEQ_LEN  1024

typedef float v2f __attribute__((ext_vector_type(2)));
typedef float v8f __attribute__((ext_vector_type(8)));

// ---------------------------------------------------------------------------
// Stateless counter-based RNG: splitmix64 finalizer over (seq, t).
// Deterministic across calls; one uniform per (sequence, step), identical on
// every lane of the wave (same inputs -> same value, no broadcast needed).
// ---------------------------------------------------------------------------
__device__ __forceinline__ float rng_u01(uint32_t seq, uint32_t t) {
    uint64_t x = ((uint64_t)seq << 32) | (uint64_t)t;
    x += 0x9E3779B97F4A7C15ull;
    x = (x ^ (x >> 30)) * 0xBF58476D1CE4E5B9ull;
    x = (x ^ (x >> 27)) * 0x94D049BB133111EBull;
    x = x ^ (x >> 31);
    // 24-bit mantissa in [0, 1)
    return (float)(uint32_t)(x >> 40) * (1.0f / 16777216.0f);
}

// ---------------------------------------------------------------------------
// Kernel 1: row-wise inclusive prefix sum (CDF) of a [n_rows x 4096] matrix,
// plus a compacted per-row block-boundary table bounds[row][32] holding
// cdf[row][128*b + 127] contiguously (one 128 B cache line per row).
//
// One 256-thread block per row. The row is staged into LDS with the CDNA5
// async global->LDS path (ASYNCcnt). Each thread scans 16 elements in
// registers; the 256 partial sums are then scanned by WAVE 0 using a chain
// of V_WMMA_F32_16X16X4_F32: with S[r][c]=sums[16r+c] and U the 16x16
// upper-triangular ones matrix, D = S x U + C (C[r][c]=row offsets) is the
// full 256-wide inclusive scan (prefix-sum-as-matmul).
// ---------------------------------------------------------------------------
__global__ __launch_bounds__(256) void row_cdf_kernel(
        const float* __restrict__ src, float* __restrict__ dst,
        float* __restrict__ bounds) {
    __shared__ float tile[K_STATES];   // 16 KB staged row
    __shared__ float sums[256];

    const int row = blockIdx.x;
    const int tid = threadIdx.x;
    const float* rp = src + (size_t)row * K_STATES;

    // --- CDNA5 async global->LDS staging (global_load_async_to_lds_b128) ---
    // Generic LDS pointers on AMDGPU carry the LDS byte offset in their low
    // 32 bits; truncation yields the offset the async op needs.
    uint32_t lds_base = (uint32_t)(uintptr_t)(&tile[0]);
#pragma unroll
    for (int i = 0; i < 4; ++i) {
        uint32_t off = (uint32_t)(tid * 16 + i * 4096);   // bytes
        uint64_t ga  = (uint64_t)(uintptr_t)rp + off;
        uint32_t la  = lds_base + off;
        asm volatile("global_load_async_to_lds_b128 %0, %1, off"
                     :: "v"(la), "v"(ga) : "memory");
    }
    asm volatile("s_wait_asynccnt 0" ::: "memory");
    __syncthreads();

    // --- per-thread inclusive prefix over 16 contiguous elements ---
    float v[16];
    float run = 0.0f;
#pragma unroll
    for (int i = 0; i < 16; ++i) {
        run += tile[tid * 16 + i];
        v[i] = run;
    }
    sums[tid] = run;
    __syncthreads();

    // --- wave 0: 256-wide inclusive scan of sums[] via WMMA ---------------
    // D = S x U + C, chained over 4 K-slices of V_WMMA_F32_16X16X4_F32.
    // Layouts (ISA 7.12.2, f32):
    //   A 16x4 : 2 VGPRs; m = lane&15; half-wave picks K pair {0,1}/{2,3}
    //   B 4x16 : 2 VGPRs; c = lane&15; half-wave picks K pair (mirrors A)
    //   C/D    : 8 VGPRs; acc[j] <-> (M = j + 8*(lane>>4), N = lane&15)
    if (tid < 32) {
        const int lane = tid;
        const int nh   = lane >> 4;    // half-wave select
        const int c    = lane & 15;    // column index (also A's row m)

        // Exclusive row offsets off[r] = sum of all elements in rows < r.
        // Broadcast LDS reads (all lanes same address stream) -> no conflicts.
        float off[16];
        {
            float accum = 0.0f;
#pragma unroll
            for (int r = 0; r < 16; ++r) {
                off[r] = accum;
                float t = 0.0f;
#pragma unroll
                for (int cc = 0; cc < 16; ++cc) t += sums[r * 16 + cc];
                accum += t;
            }
        }
        // Seed accumulator: C[r][c] = off[r] for this lane's rows.
        v8f acc;
#pragma unroll
        for (int j = 0; j < 8; ++j) acc[j] = nh ? off[8 + j] : off[j];

        // 4 chained WMMAs over K slices of S x U.
#pragma unroll
        for (int kb = 0; kb < 4; ++kb) {
            const int k0 = kb * 4 + nh * 2;          // this half's K pair
            v2f a, b;
            a.x = sums[c * 16 + k0];                 // S[m][k0],   m = c
            a.y = sums[c * 16 + k0 + 1];             // S[m][k0+1]
            b.x = (k0     <= c) ? 1.0f : 0.0f;       // U[k0][c]
            b.y = (k0 + 1 <= c) ? 1.0f : 0.0f;       // U[k0+1][c]
            acc = __builtin_amdgcn_wmma_f32_16x16x4_f32(
                      /*neg_a=*/false, a, /*neg_b=*/false, b,
                      /*c_mod=*/(short)0, acc,
                      /*reuse_a=*/false, /*reuse_b=*/false);
        }

        // Write the inclusive scan back over sums[].
#pragma unroll
        for (int j = 0; j < 8; ++j)
            sums[(nh * 8 + j) * 16 + c] = acc[j];
    }
    __syncthreads();

    const float excl = sums[tid] - run;   // exclusive prefix for this thread

    float* dp = dst + (size_t)row * K_STATES + tid * 16;
#pragma unroll
    for (int i = 0; i < 16; ++i) dp[i] = v[i] + excl;

    // Compacted block boundaries: thread 8b+7 holds cdf[row][128b+127].
    if ((tid & 7) == 7)
        bounds[(size_t)row * 32 + (tid >> 3)] = v[15] + excl;
}

// ---------------------------------------------------------------------------
// Wave-cooperative inverse-CDF search: first index with cdf[idx] >= u.
// Round 1: one contiguous 128 B line of 32 precomputed block boundaries
//          (1 L2 line for the whole wave) -> ballot picks the 128-wide block.
// Round 2: 32 lanes cover the winning 512 B block with one float4 each
//          (4 L2 lines) -> ballot + ds_bpermute give the exact index.
// 2 dependent L2 round-trips, 5 cache lines, ~640 B per step.
// ---------------------------------------------------------------------------
__device__ __forceinline__ int cdf_search(const float* __restrict__ row,
                                          const float* __restrict__ brow,
                                          float u, int lane) {
    // Round 1: pick the 128-wide block from the compacted boundary line.
    float b = brow[lane];
    uint32_t m1 = __builtin_amdgcn_ballot_w32(b >= u);
    int blk = m1 ? __builtin_ctz(m1) : 31;      // fallback: last block
    int base = blk * 128;

    // Round 2: 512 B block, one float4 per lane (16 B aligned).
    const float4 q = *(const float4*)(row + base + lane * 4);
    int k = 4;
    if      (q.x >= u) k = 0;
    else if (q.y >= u) k = 1;
    else if (q.z >= u) k = 2;
    else if (q.w >= u) k = 3;
    uint32_t m2 = __builtin_amdgcn_ballot_w32(k < 4);
    if (m2) {
        int l2 = __builtin_ctz(m2);
        int kk = __builtin_amdgcn_ds_bpermute(l2 << 2, k);  // broadcast lane l2's k
        return base + l2 * 4 + kk;
    }
    return base + 127;   // numerical-tail fallback (cdf top ~ 1.0)
}

// ---------------------------------------------------------------------------
// Kernel 2: autoregressive sampling. One wave32 per sequence, 8 waves/block.
// The 64 MB CDF + 0.5 MB boundary table stay resident in the 192 MB L2; the
// 32 MB int32 output is streamed with non-temporal stores to avoid evictions.
// ---------------------------------------------------------------------------
__global__ __launch_bounds__(256) void markov_sample_kernel(
        const float* __restrict__ cdf,         // [K_STATES, K_STATES] row CDFs
        const float* __restrict__ bounds,      // [K_STATES, 32] block tails
        const float* __restrict__ init_cdf,    // [K_STATES]
        const float* __restrict__ init_bounds, // [32]
        int* __restrict__ out) {               // [N_SEQ, SEQ_LEN]
    const int lane = threadIdx.x & 31;
    const int seq  = (blockIdx.x * 256 + threadIdx.x) >> 5;
    if (seq >= N_SEQ) return;

    int* seq_out = out + (size_t)seq * SEQ_LEN;

    float u = rng_u01((uint32_t)seq, 0u);
    int s = cdf_search(init_cdf, init_bounds, u, lane);
    if (lane == 0) __builtin_nontemporal_store(s, seq_out);

    for (int t = 1; t < SEQ_LEN; ++t) {
        u = rng_u01((uint32_t)seq, (uint32_t)t);
        s = cdf_search(cdf + (size_t)s * K_STATES,
                       bounds + (size_t)s * 32, u, lane);
        if (lane == 0) __builtin_nontemporal_store(s, seq_out + t);
    }
}

// ---------------------------------------------------------------------------
// kernel_launch
//   d_in[0] = initial_probs      (float, 4096)
//   d_in[1] = transition_probs   (float, 4096*4096)
//   d_in[2] = num_sequences      (int, 1)   [fixed 8192]
//   d_in[3] = sequence_length    (int, 1)   [fixed 1024]
//   d_out   = int32 [8192, 1024]
//   d_ws    = 64 MB transition CDF + 16 KB init CDF + 512 KB boundary tables
// ---------------------------------------------------------------------------
extern "C" void kernel_launch(void* const* d_in, const int* in_sizes, int n_in,
                              void* d_out, int out_size, void* d_ws, size_t ws_size,
                              hipStream_t stream) {
    (void)in_sizes; (void)n_in; (void)out_size; (void)ws_size;

    const float* init_probs  = (const float*)d_in[0];
    const float* trans_probs = (const float*)d_in[1];

    float* cdf         = (float*)d_ws;                                // 64 MB
    float* init_cdf    = cdf + (size_t)K_STATES * (size_t)K_STATES;   // 16 KB
    float* bounds      = init_cdf + K_STATES;                         // 512 KB
    float* init_bounds = bounds + (size_t)K_STATES * 32;              // 128 B
    int*   out         = (int*)d_out;

    // Precompute CDFs + compacted block-boundary tables (deterministic).
    row_cdf_kernel<<<K_STATES, 256, 0, stream>>>(trans_probs, cdf, bounds);
    row_cdf_kernel<<<1,        256, 0, stream>>>(init_probs, init_cdf, init_bounds);

    // 8192 sequences, 1 wave32 each, 8 waves per 256-thread block.
    markov_sample_kernel<<<N_SEQ / 8, 256, 0, stream>>>(
        cdf, bounds, init_cdf, init_bounds, out);
}